// SimplifiedDNC_77464030151156
// MI455X (gfx1250) — compile-verified
//
#include <hip/hip_runtime.h>
#include <math.h>

#define B 32
#define S 128
#define I 256
#define H 512
#define G (4*H)       // 2048 gates
#define N 2048
#define WD 64
#define R 4
#define OUT 256
#define INTF 453
#define INTF_PAD 464  // 29 tiles of 16
#define FIN (H + R*WD) // 768
#define CLIPV 20.0f
#define EPSV 1e-6f

#define XH_STRIDE 772  // 768 + 4 pad: rows land on distinct LDS bank pairs
#define Z_STRIDE 516   // 512 + 4 pad

typedef float v2f __attribute__((ext_vector_type(2)));
typedef float v8f __attribute__((ext_vector_type(8)));

__device__ __forceinline__ float sigmoidf(float x) { return 1.0f / (1.0f + expf(-x)); }
__device__ __forceinline__ float softplusf(float x) { return (x > 20.0f) ? x : log1pf(expf(x)); }

// D = A(16x4 f32) x B(4x16 f32) + C, fp32 WMMA (exact precision)
__device__ __forceinline__ v8f wmma4(v2f a, v2f b, v8f c) {
  return __builtin_amdgcn_wmma_f32_16x16x4_f32(false, a, false, b, (short)0, c, false, false);
}

// ---------------- gates = [x_t | h] @ [W_ih | W_hh]^T  (32 x 2048) ----------------
// grid 16 blocks x 256 threads (8 waves). A panel (32x768) staged once in LDS,
// shared by all 8 waves; each wave computes one N-tile for both M-tiles.
__global__ void k_gates(const float* __restrict__ xin, int t,
                        const float* __restrict__ h,
                        const float* __restrict__ Wih,
                        const float* __restrict__ Whh,
                        float* __restrict__ gates) {
  __shared__ float xh[B * XH_STRIDE];   // ~96.5 KB of the WGP's 320 KB LDS
  int tid = threadIdx.x;
  // stage x_t (32x256) : float4 coalesced
  for (int i = tid; i < B * (I / 4); i += 256) {
    int row = i >> 6, c4 = (i & 63) * 4;
    float4 v = *(const float4*)(xin + (size_t)row * (S * I) + (size_t)t * I + c4);
    *(float4*)&xh[row * XH_STRIDE + c4] = v;
  }
  // stage h (32x512)
  for (int i = tid; i < B * (H / 4); i += 256) {
    int row = i >> 7, c4 = (i & 127) * 4;
    float4 v = *(const float4*)(h + (size_t)row * H + c4);
    *(float4*)&xh[row * XH_STRIDE + I + c4] = v;
  }
  __syncthreads();

  int wave = tid >> 5, lane = tid & 31;
  int nt = blockIdx.x * 8 + wave;        // 0..127
  int n0 = nt * 16;
  int half = lane >> 4, q = lane & 15;
  int bn = n0 + q;
  v8f acc0 = {}; v8f acc1 = {};
  const float* a0p = &xh[q * XH_STRIDE];
  const float* a1p = &xh[(16 + q) * XH_STRIDE];
  const float* wi = Wih + (size_t)bn * I;
#pragma unroll 4
  for (int k = 0; k < I; k += 4) {             // K segment 1: x_t
    int kk = k + 2 * half;
    v2f a0 = *(const v2f*)(a0p + kk);          // ds_load_b64
    v2f a1 = *(const v2f*)(a1p + kk);
    v2f bb = *(const v2f*)(wi + kk);           // global_load_b64
    acc0 = wmma4(a0, bb, acc0);
    acc1 = wmma4(a1, bb, acc1);
  }
  const float* wh = Whh + (size_t)bn * H;
#pragma unroll 4
  for (int k = 0; k < H; k += 4) {             // K segment 2: h
    int kk = k + 2 * half;
    v2f a0 = *(const v2f*)(a0p + I + kk);
    v2f a1 = *(const v2f*)(a1p + I + kk);
    v2f bb = *(const v2f*)(wh + kk);
    acc0 = wmma4(a0, bb, acc0);
    acc1 = wmma4(a1, bb, acc1);
  }
#pragma unroll
  for (int v = 0; v < 8; v++) {                // D: VGPR v -> M=v (lanes0-15), M=v+8 (lanes16-31)
    int row = v + 8 * half;
    gates[(size_t)row * G + n0 + q]        = acc0[v];
    gates[(size_t)(16 + row) * G + n0 + q] = acc1[v];
  }
}

// ---------------- LSTM elementwise + clip + LayerNorm -> z ----------------
__global__ void k_lstm_ln(const float* __restrict__ gates,
                          const float* __restrict__ bih, const float* __restrict__ bhh,
                          float* __restrict__ h, float* __restrict__ c,
                          const float* __restrict__ lng, const float* __restrict__ lnb,
                          float* __restrict__ z) {
  __shared__ float red[512];
  int b = blockIdx.x, j = threadIdx.x;
  const float* gr = gates + (size_t)b * G;
  float gi = gr[j]         + bih[j]         + bhh[j];
  float gf = gr[H + j]     + bih[H + j]     + bhh[H + j];
  float gg = gr[2 * H + j] + bih[2 * H + j] + bhh[2 * H + j];
  float go = gr[3 * H + j] + bih[3 * H + j] + bhh[3 * H + j];
  float cv = sigmoidf(gf) * c[(size_t)b * H + j] + sigmoidf(gi) * tanhf(gg);
  float hv = sigmoidf(go) * tanhf(cv);
  c[(size_t)b * H + j] = cv;
  h[(size_t)b * H + j] = hv;
  float hc = fminf(fmaxf(hv, -CLIPV), CLIPV);
  red[j] = hc; __syncthreads();
  for (int s = 256; s > 0; s >>= 1) { if (j < s) red[j] += red[j + s]; __syncthreads(); }
  float mu = red[0] * (1.0f / H); __syncthreads();
  float d = hc - mu;
  red[j] = d * d; __syncthreads();
  for (int s = 256; s > 0; s >>= 1) { if (j < s) red[j] += red[j + s]; __syncthreads(); }
  float var = red[0] * (1.0f / H);
  z[(size_t)b * H + j] = d * rsqrtf(var + 1e-5f) * lng[j] + lnb[j];
}

// ---------------- itf = z @ W_int^T  (32 x 453, padded stride 464) ----------------
__global__ void k_itf(const float* __restrict__ z, const float* __restrict__ Wint,
                      float* __restrict__ itf) {
  __shared__ float zs[B * Z_STRIDE];    // 66 KB
  int tid = threadIdx.x;
  for (int i = tid; i < B * (H / 4); i += 256) {
    int row = i >> 7, c4 = (i & 127) * 4;
    float4 v = *(const float4*)(z + (size_t)row * H + c4);
    *(float4*)&zs[row * Z_STRIDE + c4] = v;
  }
  __syncthreads();

  int wave = tid >> 5, lane = tid & 31;
  int nt = blockIdx.x * 8 + wave;
  if (nt < 29) {                         // wave-uniform guard: EXEC all-ones inside
    int n0 = nt * 16;
    int half = lane >> 4, q = lane & 15;
    int bn = n0 + q;
    int bnc = bn < INTF ? bn : (INTF - 1);  // clamp OOB rows (cols >=453 never read)
    v8f acc0 = {}; v8f acc1 = {};
    const float* a0p = &zs[q * Z_STRIDE];
    const float* a1p = &zs[(16 + q) * Z_STRIDE];
    const float* wr = Wint + (size_t)bnc * H;
#pragma unroll 4
    for (int k = 0; k < H; k += 4) {
      int kk = k + 2 * half;
      v2f a0 = *(const v2f*)(a0p + kk);
      v2f a1 = *(const v2f*)(a1p + kk);
      v2f bb = *(const v2f*)(wr + kk);
      acc0 = wmma4(a0, bb, acc0);
      acc1 = wmma4(a1, bb, acc1);
    }
#pragma unroll
    for (int v = 0; v < 8; v++) {
      int row = v + 8 * half;
      itf[(size_t)row * INTF_PAD + n0 + q]        = acc0[v];
      itf[(size_t)(16 + row) * INTF_PAD + n0 + q] = acc1[v];
    }
  }
}

// ---------------- write head: cosine softmax over N, memory update, post norms ----------------
__global__ void k_write(const float* __restrict__ itf, const float* __restrict__ bint,
                        float* __restrict__ M, float* __restrict__ whist,
                        float* __restrict__ mnorm, int t) {
  __shared__ float skey[WD], sera[WD], swv[WD];
  __shared__ float red[256];
  int b = blockIdx.x, tid = threadIdx.x;
  const float* row = itf + (size_t)b * INTF_PAD;
  if (tid < WD) {
    skey[tid] = row[260 + tid] + bint[260 + tid];                 // write_key
    sera[tid] = sigmoidf(row[325 + tid] + bint[325 + tid]);       // erase
    swv[tid]  = row[389 + tid] + bint[389 + tid];                 // write_vec
  }
  __syncthreads();
  float wstr = 1.0f + softplusf(row[324] + bint[324]);
  float kss = EPSV;
  for (int j = 0; j < WD; j++) kss += skey[j] * skey[j];
  float krn = rsqrtf(kss);
  float* Mb = M + (size_t)b * N * WD;
  float l[8]; float lmax = -1e30f;
  for (int s = 0; s < 8; s++) {
    int n = tid + s * 256;
    const float* mr = Mb + (size_t)n * WD;
    float ss = EPSV, dot = 0.f;
    for (int j = 0; j < WD; j++) { float m = mr[j]; ss += m * m; dot += m * skey[j]; }
    l[s] = wstr * dot * rsqrtf(ss) * krn;
    lmax = fmaxf(lmax, l[s]);
  }
  red[tid] = lmax; __syncthreads();
  for (int s2 = 128; s2 > 0; s2 >>= 1) { if (tid < s2) red[tid] = fmaxf(red[tid], red[tid + s2]); __syncthreads(); }
  float mx = red[0]; __syncthreads();
  float lsum = 0.f;
  for (int s = 0; s < 8; s++) { l[s] = expf(l[s] - mx); lsum += l[s]; }
  red[tid] = lsum; __syncthreads();
  for (int s2 = 128; s2 > 0; s2 >>= 1) { if (tid < s2) red[tid] += red[tid + s2]; __syncthreads(); }
  float inv = 1.0f / red[0];
  float* wh = whist + (size_t)b * S * N + (size_t)t * N;
  for (int s = 0; s < 8; s++) {
    int n = tid + s * 256;
    float w = l[s] * inv;
    wh[n] = w;
    float* mr = Mb + (size_t)n * WD;
    float ss = EPSV;
    for (int j = 0; j < WD; j++) {
      float m = mr[j];
      m = m * (1.0f - w * sera[j]) + w * swv[j];
      mr[j] = m;
      ss += m * m;
    }
    mnorm[(size_t)b * N + n] = rsqrtf(ss);   // post-update row norms for read heads
  }
}

// ---------------- read heads: 4 fused cosine softmaxes + read vectors ----------------
__global__ void k_read(const float* __restrict__ itf, const float* __restrict__ bint,
                       const float* __restrict__ M, const float* __restrict__ mnorm,
                       float* __restrict__ rhist, float* __restrict__ readvec, int t) {
  __shared__ float sw[R * N];        // 32 KB logits->weights
  __shared__ float skey[R * WD];     // 1 KB
  __shared__ float red[256];
  __shared__ float sacc[4 * R * WD]; // 4 KB
  int b = blockIdx.x, tid = threadIdx.x;
  const float* row = itf + (size_t)b * INTF_PAD;
  skey[tid] = row[tid] + bint[tid];  // read keys: itf[0..256)
  __syncthreads();
  float rstr[R], krn[R];
#pragma unroll
  for (int r = 0; r < R; r++) {
    rstr[r] = 1.0f + softplusf(row[256 + r] + bint[256 + r]);
    float ss = EPSV;
    for (int j = 0; j < WD; j++) { float k = skey[r * WD + j]; ss += k * k; }
    krn[r] = rsqrtf(ss);
  }
  const float* Mb = M + (size_t)b * N * WD;
  const float* nb = mnorm + (size_t)b * N;
  for (int s = 0; s < 8; s++) {
    int n = tid + s * 256;
    const float* mr = Mb + (size_t)n * WD;
    float mn = nb[n];
    float d0 = 0, d1 = 0, d2 = 0, d3 = 0;
    for (int j = 0; j < WD; j++) {
      float m = mr[j];
      d0 += m * skey[0 * WD + j]; d1 += m * skey[1 * WD + j];
      d2 += m * skey[2 * WD + j]; d3 += m * skey[3 * WD + j];
    }
    sw[0 * N + n] = rstr[0] * d0 * mn * krn[0];
    sw[1 * N + n] = rstr[1] * d1 * mn * krn[1];
    sw[2 * N + n] = rstr[2] * d2 * mn * krn[2];
    sw[3 * N + n] = rstr[3] * d3 * mn * krn[3];
  }
  __syncthreads();
  float* rh = rhist + (size_t)b * S * R * N + (size_t)t * R * N;
  for (int r = 0; r < R; r++) {
    float lmax = -1e30f;
    for (int s = 0; s < 8; s++) lmax = fmaxf(lmax, sw[r * N + tid + s * 256]);
    red[tid] = lmax; __syncthreads();
    for (int s2 = 128; s2 > 0; s2 >>= 1) { if (tid < s2) red[tid] = fmaxf(red[tid], red[tid + s2]); __syncthreads(); }
    float mx = red[0]; __syncthreads();
    float e[8]; float lsum = 0.f;
    for (int s = 0; s < 8; s++) { e[s] = expf(sw[r * N + tid + s * 256] - mx); lsum += e[s]; }
    red[tid] = lsum; __syncthreads();
    for (int s2 = 128; s2 > 0; s2 >>= 1) { if (tid < s2) red[tid] += red[tid + s2]; __syncthreads(); }
    float inv = 1.0f / red[0]; __syncthreads();
    for (int s = 0; s < 8; s++) {
      int n = tid + s * 256;
      float w = e[s] * inv;
      sw[r * N + n] = w;
      rh[(size_t)r * N + n] = w;
    }
    __syncthreads();
  }
  // read[b,r,:] = sum_n w[r,n] * M[b,n,:]  — 4 slot-groups of 512, j = column
  int g = tid >> 6, j = tid & 63;
  float a0 = 0, a1 = 0, a2 = 0, a3 = 0;
  int nbeg = g * 512, nend = nbeg + 512;
  for (int n = nbeg; n < nend; n++) {
    float m = Mb[(size_t)n * WD + j];   // coalesced over j; sw reads broadcast
    a0 += sw[0 * N + n] * m; a1 += sw[1 * N + n] * m;
    a2 += sw[2 * N + n] * m; a3 += sw[3 * N + n] * m;
  }
  sacc[g * R * WD + 0 * WD + j] = a0;
  sacc[g * R * WD + 1 * WD + j] = a1;
  sacc[g * R * WD + 2 * WD + j] = a2;
  sacc[g * R * WD + 3 * WD + j] = a3;
  __syncthreads();
  {
    int r = tid >> 6, jj = tid & 63;
    float sum = 0.f;
    for (int gg = 0; gg < 4; gg++) sum += sacc[gg * R * WD + r * WD + jj];
    readvec[(size_t)b * (R * WD) + r * WD + jj] = sum;
  }
}

// ---------------- out = [h | read] @ W_out^T + b_out  (last step only) ----------------
__global__ void k_out(const float* __restrict__ h, const float* __restrict__ readvec,
                      const float* __restrict__ Wout, const float* __restrict__ bout,
                      float* __restrict__ out) {
  __shared__ float xh[B * XH_STRIDE];
  int tid = threadIdx.x;
  for (int i = tid; i < B * (H / 4); i += 256) {
    int row = i >> 7, c4 = (i & 127) * 4;
    float4 v = *(const float4*)(h + (size_t)row * H + c4);
    *(float4*)&xh[row * XH_STRIDE + c4] = v;
  }
  for (int i = tid; i < B * (R * WD / 4); i += 256) {
    int row = i >> 6, c4 = (i & 63) * 4;
    float4 v = *(const float4*)(readvec + (size_t)row * (R * WD) + c4);
    *(float4*)&xh[row * XH_STRIDE + H + c4] = v;
  }
  __syncthreads();

  int wave = tid >> 5, lane = tid & 31;
  int nt = blockIdx.x * 8 + wave;  // 0..15
  int n0 = nt * 16;
  int half = lane >> 4, q = lane & 15;
  int bn = n0 + q;
  v8f acc0 = {}; v8f acc1 = {};
  const float* a0p = &xh[q * XH_STRIDE];
  const float* a1p = &xh[(16 + q) * XH_STRIDE];
  const float* w0 = Wout + (size_t)bn * FIN;
#pragma unroll 4
  for (int k = 0; k < FIN; k += 4) {
    int kk = k + 2 * half;
    v2f a0 = *(const v2f*)(a0p + kk);
    v2f a1 = *(const v2f*)(a1p + kk);
    v2f bb = *(const v2f*)(w0 + kk);
    acc0 = wmma4(a0, bb, acc0);
    acc1 = wmma4(a1, bb, acc1);
  }
  float bo = bout[n0 + q];
#pragma unroll
  for (int v = 0; v < 8; v++) {
    int rrow = v + 8 * half;
    out[(size_t)rrow * OUT + n0 + q]        = acc0[v] + bo;
    out[(size_t)(16 + rrow) * OUT + n0 + q] = acc1[v] + bo;
  }
}

extern "C" void kernel_launch(void* const* d_in, const int* in_sizes, int n_in,
                              void* d_out, int out_size, void* d_ws, size_t ws_size,
                              hipStream_t stream) {
  (void)in_sizes; (void)n_in; (void)out_size; (void)ws_size;
  const float* inputs = (const float*)d_in[0];
  const float* h0   = (const float*)d_in[1];
  const float* c0   = (const float*)d_in[2];
  const float* mem0 = (const float*)d_in[3];
  const float* Wih  = (const float*)d_in[4];
  const float* Whh  = (const float*)d_in[5];
  const float* bih  = (const float*)d_in[6];
  const float* bhh  = (const float*)d_in[7];
  const float* lng  = (const float*)d_in[8];
  const float* lnb  = (const float*)d_in[9];
  const float* Wint = (const float*)d_in[10];
  const float* bint = (const float*)d_in[11];
  const float* Wout = (const float*)d_in[12];
  const float* bout = (const float*)d_in[13];

  // d_out layout: out(32x256), h(32x512), c(32x512), M(32x2048x64),
  //               read_hist(32x128x4x2048), write_hist(32x128x2048)
  float* out   = (float*)d_out;
  float* h     = out + B * OUT;
  float* c     = h + B * H;
  float* M     = c + B * H;
  float* rhist = M + (size_t)B * N * WD;
  float* whist = rhist + (size_t)B * S * R * N;

  float* ws    = (float*)d_ws;
  float* gates = ws;                         // B*G
  float* z     = gates + B * G;              // B*H
  float* itf   = z + B * H;                  // B*INTF_PAD
  float* mnorm = itf + B * INTF_PAD;         // B*N
  float* readv = mnorm + (size_t)B * N;      // B*R*WD

  // live state lives directly in d_out regions
  hipMemcpyAsync(h, h0, (size_t)B * H * sizeof(float), hipMemcpyDeviceToDevice, stream);
  hipMemcpyAsync(c, c0, (size_t)B * H * sizeof(float), hipMemcpyDeviceToDevice, stream);
  hipMemcpyAsync(M, mem0, (size_t)B * N * WD * sizeof(float), hipMemcpyDeviceToDevice, stream);

  for (int t = 0; t < S; t++) {
    k_gates<<<16, 256, 0, stream>>>(inputs, t, h, Wih, Whh, gates);
    k_lstm_ln<<<B, 512, 0, stream>>>(gates, bih, bhh, h, c, lng, lnb, z);
    k_itf<<<4, 256, 0, stream>>>(z, Wint, itf);
    k_write<<<B, 256, 0, stream>>>(itf, bint, M, whist, mnorm, t);
    k_read<<<B, 256, 0, stream>>>(itf, bint, M, mnorm, rhist, readv, t);
  }
  k_out<<<2, 256, 0, stream>>>(h, readv, Wout, bout, out);
}